// EncoderLayer_30408368456098
// MI455X (gfx1250) — compile-verified
//
#include <hip/hip_runtime.h>
#include <cstdint>

typedef __attribute__((ext_vector_type(2))) float v2f;
typedef __attribute__((ext_vector_type(8))) float v8f;

#define B_SZ   8
#define N_SEQ  512
#define D_DIM  1024
#define S_SL   16
#define H_HD   8
#define DFF2   2048
#define M_ROWS (B_SZ * N_SEQ)   // 4096

__device__ __forceinline__ v8f wmma4(v2f a, v2f b, v8f c) {
  // D = A(16x4 f32) * B(4x16 f32) + C(16x16 f32); exact f32 accumulate
  return __builtin_amdgcn_wmma_f32_16x16x4_f32(false, a, false, b, (short)0, c,
                                               false, false);
}

// -------------------------------------------------------------------------
// Generic GEMM: C[M x N] = A[M x K] @ B[K x N] + bias[N], optional ReLU.
// One wave computes a 32x32 tile via 2x2 blocking of 16x16 WMMA tiles.
// grid = (N/32, M/32), block = 32 (one wave). All dims divide 32/4 here.
// -------------------------------------------------------------------------
__global__ __launch_bounds__(32) void gemm_bias_wmma(
    const float* __restrict__ A, const float* __restrict__ Bm,
    const float* __restrict__ bias, float* __restrict__ C,
    int K, int lda, int ldb, int ldc, int relu)
{
  const int lane = threadIdx.x & 31;
  const int half = lane >> 4;      // 0: K pair {0,1}, 1: K pair {2,3}
  const int l15  = lane & 15;
  const int m0   = blockIdx.y << 5;
  const int n0   = blockIdx.x << 5;

  v8f acc00 = {}; v8f acc01 = {}; v8f acc10 = {}; v8f acc11 = {};

  const float* Ar0 = A + (size_t)(m0 + l15) * lda + 2 * half;
  const float* Ar1 = Ar0 + (size_t)16 * lda;
  const float* Bp  = Bm + (size_t)(2 * half) * ldb;

  #pragma unroll 4
  for (int k0 = 0; k0 < K; k0 += 4) {
    v2f a0 = *(const v2f*)(Ar0 + k0);              // A[m][k0+2h .. +1]
    v2f a1 = *(const v2f*)(Ar1 + k0);
    const float* bk = Bp + (size_t)k0 * ldb;       // row k0+2h of B
    v2f bf0, bf1;
    bf0.x = bk[n0 + l15];
    bf0.y = bk[(size_t)ldb + n0 + l15];
    bf1.x = bk[n0 + 16 + l15];
    bf1.y = bk[(size_t)ldb + n0 + 16 + l15];
    acc00 = wmma4(a0, bf0, acc00);
    acc01 = wmma4(a0, bf1, acc01);
    acc10 = wmma4(a1, bf0, acc10);
    acc11 = wmma4(a1, bf1, acc11);
  }

  const float bi0 = bias[n0 + l15];
  const float bi1 = bias[n0 + 16 + l15];
  #pragma unroll
  for (int vv = 0; vv < 8; ++vv) {
    int r = vv + 8 * half;
    float o00 = acc00[vv] + bi0;
    float o01 = acc01[vv] + bi1;
    float o10 = acc10[vv] + bi0;
    float o11 = acc11[vv] + bi1;
    if (relu) {
      o00 = fmaxf(o00, 0.f); o01 = fmaxf(o01, 0.f);
      o10 = fmaxf(o10, 0.f); o11 = fmaxf(o11, 0.f);
    }
    float* c0 = C + (size_t)(m0 + r) * ldc;
    float* c1 = C + (size_t)(m0 + 16 + r) * ldc;
    c0[n0 + l15]      = o00;
    c0[n0 + 16 + l15] = o01;
    c1[n0 + l15]      = o10;
    c1[n0 + 16 + l15] = o11;
  }
}

// -------------------------------------------------------------------------
// Fused per-head attention. One wave per (head, quarter-of-rows).
// head = (s,b,h); Q/K/V slice lives at column offset h*128 + s*8 (DH=8).
// K,V staged in LDS; scores kept as unnormalized exp in padded LDS tile;
// both score and context matmuls use f32 WMMA. Writes ctx in (S,B,N,64).
// -------------------------------------------------------------------------
#define P_STRIDE 516  // 512 + 4 pad -> avoids 16-way LDS bank conflicts

__global__ __launch_bounds__(32) void attn_head_wmma(
    const float* __restrict__ q, const float* __restrict__ k,
    const float* __restrict__ v, const float* __restrict__ mask,
    float* __restrict__ ctx)
{
  extern __shared__ float smem[];
  float* Ksh = smem;           // [512][8]
  float* Vsh = smem + 4096;    // [512][8]
  float* P   = smem + 8192;    // [16][P_STRIDE]

  const int hid = blockIdx.x;          // 0 .. S*B*H-1
  const int s   = hid >> 6;            // / (B*H)
  const int rem = hid & 63;
  const int b   = rem >> 3;
  const int h   = rem & 7;

  const int lane = threadIdx.x & 31;
  const int half = lane >> 4;
  const int l15  = lane & 15;
  const size_t headoff = (size_t)h * 128 + (size_t)s * 8;

  // Stage K and V (512 x 8 each) into LDS
  for (int n = lane; n < N_SEQ; n += 32) {
    const float* kr = k + ((size_t)(b * N_SEQ + n)) * D_DIM + headoff;
    const float* vr = v + ((size_t)(b * N_SEQ + n)) * D_DIM + headoff;
    *(float4*)(Ksh + n * 8)     = *(const float4*)(kr);
    *(float4*)(Ksh + n * 8 + 4) = *(const float4*)(kr + 4);
    *(float4*)(Vsh + n * 8)     = *(const float4*)(vr);
    *(float4*)(Vsh + n * 8 + 4) = *(const float4*)(vr + 4);
  }
  __syncthreads();

  const float scale = 0.35355339059327373f;  // 1/sqrt(8)

  for (int qb = blockIdx.y * 8; qb < blockIdx.y * 8 + 8; ++qb) {
    const int qrow0 = qb * 16;
    // Q fragments for this 16-row block (K dim = 8 -> two k-steps)
    const float* qr = q + ((size_t)(b * N_SEQ + qrow0 + l15)) * D_DIM + headoff;
    v2f qa0 = *(const v2f*)(qr + 2 * half);       // K 0..3
    v2f qa1 = *(const v2f*)(qr + 4 + 2 * half);   // K 4..7

    // ---- scores = Q K^T * scale + mask * -1e9, stored to LDS ----
    for (int t = 0; t < 32; ++t) {
      const int n0 = t * 16;
      v8f acc = {};
      const float* kp = Ksh + (size_t)(n0 + l15) * 8 + 2 * half;
      v2f bf0 = *(const v2f*)(kp);       // B[kd][key] = K[key][kd], kd 0..3
      v2f bf1 = *(const v2f*)(kp + 4);   // kd 4..7
      acc = wmma4(qa0, bf0, acc);
      acc = wmma4(qa1, bf1, acc);
      #pragma unroll
      for (int vv = 0; vv < 8; ++vv) {
        int r = vv + 8 * half;
        float m = mask[((size_t)b * N_SEQ + qrow0 + r) * N_SEQ + n0 + l15];
        P[r * P_STRIDE + n0 + l15] = acc[vv] * scale + m * (-1e9f);
      }
    }
    __syncthreads();

    // ---- softmax (rows split across the two wave halves) ----
    float* Prow = P + l15 * P_STRIDE + half * 256;
    float mx = -3.4e38f;
    #pragma unroll 8
    for (int c = 0; c < 256; ++c) mx = fmaxf(mx, Prow[c]);
    mx = fmaxf(mx, __shfl_xor(mx, 16));
    float sum = 0.f;
    #pragma unroll 8
    for (int c = 0; c < 256; ++c) {
      float e = __expf(Prow[c] - mx);
      Prow[c] = e;
      sum += e;
    }
    sum += __shfl_xor(sum, 16);
    float inv = 1.0f / sum;
    __syncthreads();

    // ---- ctx = P V  (V padded from 8 to 16 cols with zeros) ----
    v8f acc = {};
    #pragma unroll 4
    for (int k0 = 0; k0 < N_SEQ; k0 += 4) {
      v2f a = *(const v2f*)(P + (size_t)l15 * P_STRIDE + k0 + 2 * half);
      int kd = k0 + 2 * half;
      v2f bb;
      bb.x = (l15 < 8) ? Vsh[(size_t)kd * 8 + l15]       : 0.f;
      bb.y = (l15 < 8) ? Vsh[(size_t)(kd + 1) * 8 + l15] : 0.f;
      acc = wmma4(a, bb, acc);
    }
    #pragma unroll
    for (int vv = 0; vv < 8; ++vv) {
      int r = vv + 8 * half;
      float invr = __shfl(inv, r);   // lane r holds 1/rowsum for row r
      if (l15 < 8) {
        ctx[(((size_t)s * B_SZ + b) * N_SEQ + qrow0 + r) * 64 + h * 8 + l15] =
            acc[vv] * invr;
      }
    }
    __syncthreads();
  }
}

// -------------------------------------------------------------------------
// out = LayerNorm(x + y) over last dim (1024), affine g,b. One block per row.
// -------------------------------------------------------------------------
__global__ __launch_bounds__(256) void add_ln_kernel(
    const float* __restrict__ x, const float* __restrict__ y,
    const float* __restrict__ g, const float* __restrict__ bb,
    float* __restrict__ out)
{
  const int row = blockIdx.x;
  const int t   = threadIdx.x;
  const float* xr = x + (size_t)row * D_DIM;
  const float* yr = y + (size_t)row * D_DIM;

  float4 xv = *(const float4*)(xr + t * 4);
  float4 yv = *(const float4*)(yr + t * 4);
  float4 sv = make_float4(xv.x + yv.x, xv.y + yv.y, xv.z + yv.z, xv.w + yv.w);

  __shared__ float red[256];
  float part = sv.x + sv.y + sv.z + sv.w;
  red[t] = part;
  __syncthreads();
  for (int off = 128; off > 0; off >>= 1) {
    if (t < off) red[t] += red[t + off];
    __syncthreads();
  }
  const float mu = red[0] * (1.0f / 1024.0f);
  __syncthreads();

  float dx = sv.x - mu, dy = sv.y - mu, dz = sv.z - mu, dw = sv.w - mu;
  red[t] = dx * dx + dy * dy + dz * dz + dw * dw;
  __syncthreads();
  for (int off = 128; off > 0; off >>= 1) {
    if (t < off) red[t] += red[t + off];
    __syncthreads();
  }
  const float var = red[0] * (1.0f / 1024.0f);
  const float rs  = rsqrtf(var + 1e-6f);

  float* orow = out + (size_t)row * D_DIM;
  orow[t * 4 + 0] = dx * rs * g[t * 4 + 0] + bb[t * 4 + 0];
  orow[t * 4 + 1] = dy * rs * g[t * 4 + 1] + bb[t * 4 + 1];
  orow[t * 4 + 2] = dz * rs * g[t * 4 + 2] + bb[t * 4 + 2];
  orow[t * 4 + 3] = dw * rs * g[t * 4 + 3] + bb[t * 4 + 3];
}

// -------------------------------------------------------------------------
extern "C" void kernel_launch(void* const* d_in, const int* in_sizes, int n_in,
                              void* d_out, int out_size, void* d_ws, size_t ws_size,
                              hipStream_t stream) {
  const float* x     = (const float*)d_in[0];
  const float* mask  = (const float*)d_in[1];
  const float* w_q   = (const float*)d_in[2];
  const float* b_q   = (const float*)d_in[3];
  const float* w_k   = (const float*)d_in[4];
  const float* b_k   = (const float*)d_in[5];
  const float* w_v   = (const float*)d_in[6];
  const float* b_v   = (const float*)d_in[7];
  const float* w_pi  = (const float*)d_in[8];
  const float* b_pi  = (const float*)d_in[9];
  const float* w_o   = (const float*)d_in[10];
  const float* b_o   = (const float*)d_in[11];
  const float* w_f1  = (const float*)d_in[12];
  const float* b_f1  = (const float*)d_in[13];
  const float* w_f2  = (const float*)d_in[14];
  const float* b_f2  = (const float*)d_in[15];
  const float* ln1_g = (const float*)d_in[16];
  const float* ln1_b = (const float*)d_in[17];
  const float* ln2_g = (const float*)d_in[18];
  const float* ln2_b = (const float*)d_in[19];
  float* out = (float*)d_out;

  float* ws = (float*)d_ws;
  const size_t MD = (size_t)M_ROWS * D_DIM;  // 4,194,304 floats
  float* qb     = ws;
  float* kb     = ws + 1 * MD;
  float* vb     = ws + 2 * MD;
  float* ctxb   = ws + 3 * MD;   // (S,B,N,64) == MD floats
  float* concat = ws + 4 * MD;   // (B,N,1024)
  float* tmp    = ws + 5 * MD;   // attn_out / ffn2
  float* out1   = ws + 6 * MD;
  float* ffn1   = ws + 7 * MD;   // 2*MD floats

  const dim3 blk(32);
  const dim3 gProj(D_DIM / 32, M_ROWS / 32);   // (32,128)

  // q, k, v projections
  gemm_bias_wmma<<<gProj, blk, 0, stream>>>(x, w_q, b_q, qb, D_DIM, D_DIM, D_DIM, D_DIM, 0);
  gemm_bias_wmma<<<gProj, blk, 0, stream>>>(x, w_k, b_k, kb, D_DIM, D_DIM, D_DIM, D_DIM, 0);
  gemm_bias_wmma<<<gProj, blk, 0, stream>>>(x, w_v, b_v, vb, D_DIM, D_DIM, D_DIM, D_DIM, 0);

  // fused attention over S*B*H = 1024 heads, 4 row-quarters each
  const size_t smem_bytes = (size_t)(4096 + 4096 + 16 * P_STRIDE) * sizeof(float);
  attn_head_wmma<<<dim3(S_SL * B_SZ * H_HD, 4), blk, smem_bytes, stream>>>(
      qb, kb, vb, mask, ctxb);

  // per-slice 64x64 projections; strided output implements concat directly
  for (int s = 0; s < S_SL; ++s) {
    gemm_bias_wmma<<<dim3(64 / 32, M_ROWS / 32), blk, 0, stream>>>(
        ctxb + (size_t)s * M_ROWS * 64, w_pi + (size_t)s * 64 * 64,
        b_pi + (size_t)s * 64, concat + (size_t)s * 64,
        64, 64, 64, D_DIM, 0);
  }

  // output projection
  gemm_bias_wmma<<<gProj, blk, 0, stream>>>(concat, w_o, b_o, tmp, D_DIM, D_DIM, D_DIM, D_DIM, 0);

  // out1 = LN(x + attn_out)
  add_ln_kernel<<<M_ROWS, 256, 0, stream>>>(x, tmp, ln1_g, ln1_b, out1);

  // FFN
  gemm_bias_wmma<<<dim3(DFF2 / 32, M_ROWS / 32), blk, 0, stream>>>(
      out1, w_f1, b_f1, ffn1, D_DIM, D_DIM, DFF2, DFF2, 1);
  gemm_bias_wmma<<<dim3(D_DIM / 32, M_ROWS / 32), blk, 0, stream>>>(
      ffn1, w_f2, b_f2, tmp, DFF2, DFF2, D_DIM, D_DIM, 0);

  // out = LN(out1 + ffn2)
  add_ln_kernel<<<M_ROWS, 256, 0, stream>>>(out1, tmp, ln2_g, ln2_b, out);
}